// StackMemory_9122510536894
// MI455X (gfx1250) — compile-verified
//
#include <hip/hip_runtime.h>

// ---------------------------------------------------------------------------
// StackMemory on MI455X (gfx1250, wave32).
//
// Reference reduces to:  out[t,0,h] = softmax3(h_t.W_ap + b)[0] * sigmoid(h_t.D)
//                        out[t,1:32,h] = 0
//
// Roofline: 400 MB of output stores dominate (23.3 TB/s -> ~17 us floor);
// 13 MB of reads and ~26 MFLOP of GEMV are noise.  Structure:
//   kernel 1: zero-fill rows 1..31 with non-temporal b128 stores (387.5 MB)
//   kernel 2: exact-f32 WMMA GEMV (v_wmma_f32_16x16x4_f32, K=400=100x4,
//             no tail, no precision loss, no convert VALU), 16 timesteps
//             per wave, then NT-broadcast the scalar into stack row 0.
// ---------------------------------------------------------------------------

typedef __attribute__((ext_vector_type(8))) float v8f;
typedef __attribute__((ext_vector_type(4))) float v4f;
typedef __attribute__((ext_vector_type(2))) float v2f;

#define HIDDEN 400
#define DEPTH  32
#define SEQ    8192
#define ROWF   (HIDDEN * DEPTH)   /* floats per timestep = 12800 */

// ---------------------------------------------------------------------------
// Kernel 1: zero rows 1..31 of every timestep. One block per timestep:
// 3100 float4 NT stores spread over 256 threads (~12 each), no divisions.
// ---------------------------------------------------------------------------
__global__ void zero_fill_kernel(v4f* __restrict__ out) {
    const int t     = blockIdx.x;                    // 0..8191
    const int per_t = (HIDDEN * (DEPTH - 1)) / 4;    // 3100 float4 (rows 1..31)
    v4f z; z.x = 0.f; z.y = 0.f; z.z = 0.f; z.w = 0.f;
    v4f* dst = out + (size_t)t * (ROWF / 4) + (HIDDEN / 4);  // skip row 0
    for (int i = threadIdx.x; i < per_t; i += blockDim.x)
        __builtin_nontemporal_store(z, dst + i);
}

// ---------------------------------------------------------------------------
// Kernel 2: exact-f32 WMMA GEMV + activations + row-0 broadcast.
// One wave handles 16 timesteps. A = 16x400 hidden tile (f32), B = 400x16
// packed weights [W_ap0 | W_ap1 | W_ap2 | D | 0...], C = 16x16 f32.
// ---------------------------------------------------------------------------
__global__ void stack_wmma_kernel(const float* __restrict__ hs,
                                  const float* __restrict__ W,
                                  const float* __restrict__ bias,
                                  const float* __restrict__ Dv,
                                  float* __restrict__ out) {
    const int lane = threadIdx.x & 31;
    const int wave = blockIdx.x * (blockDim.x >> 5) + (threadIdx.x >> 5);
    const int t0   = wave * 16;
    if (t0 >= SEQ) return;

    // ISA 7.12.2 32-bit A-matrix 16x4 layout:
    //   lanes 0-15 : M = lane,    VGPR0 = K=0, VGPR1 = K=1
    //   lanes 16-31: M = lane-16, VGPR0 = K=2, VGPR1 = K=3
    const int row  = lane & 15;
    const int koff = (lane >> 4) << 1;               // 0 or 2
    const float* hrow = hs + (size_t)(t0 + row) * HIDDEN;

    // B column for this lane (N = lane&15): 0..2 -> W_ap rows, 3 -> D, else 0.
    // Invalid lanes still read (in-bounds, from Dv) and are zeroed by cndmask
    // to keep control flow uniform in the hot loop.
    const int    ncol   = lane & 15;
    const bool   nvalid = (ncol < 4);
    const float* wp     = (ncol < 3) ? (W + (size_t)ncol * HIDDEN) : Dv;

    // one speculative prefetch of this lane's A row (global_prefetch_b8)
    __builtin_prefetch(hrow, 0, 0);

    v8f c = {0.f, 0.f, 0.f, 0.f, 0.f, 0.f, 0.f, 0.f};

    // K = 400 = 100 x 4: single tight loop, no tail, exact f32 accumulate.
#pragma unroll 4
    for (int kb = 0; kb < HIDDEN; kb += 4) {
        v2f a = *(const v2f*)(hrow + kb + koff);     // 8B-aligned b64 load
        v2f b = *(const v2f*)(wp   + kb + koff);     // broadcast within wave
        if (!nvalid) { b.x = 0.f; b.y = 0.f; }       // columns 4..15 are zero
        // D = A x B + C   (v_wmma_f32_16x16x4_f32)
        c = __builtin_amdgcn_wmma_f32_16x16x4_f32(
                /*neg_a=*/false, a, /*neg_b=*/false, b,
                /*c_mod=*/(short)0, c, /*reuse_a=*/false, /*reuse_b=*/false);
    }

    // C layout: VGPR r, lane L -> M = r + (L>=16 ? 8 : 0), N = L&15.
    // Gather logits (N=0..2) and D-dot (N=3) via wave32 shuffles.
    const int base = lane & 16;
    const float bb0 = bias[0], bb1 = bias[1], bb2 = bias[2];
    float sarr[8];
#pragma unroll
    for (int r = 0; r < 8; ++r) {
        float l0 = __shfl(c[r], base + 0, 32) + bb0;
        float l1 = __shfl(c[r], base + 1, 32) + bb1;
        float l2 = __shfl(c[r], base + 2, 32) + bb2;
        float dd = __shfl(c[r], base + 3, 32);
        float mx = fmaxf(l0, fmaxf(l1, l2));
        float e0 = __expf(l0 - mx);
        float e1 = __expf(l1 - mx);
        float e2 = __expf(l2 - mx);
        float push = e0 / (e0 + e1 + e2);
        float pv   = 1.0f / (1.0f + __expf(-dd));
        sarr[r] = push * pv;              // the broadcast scalar for row M
    }

    // Broadcast each timestep's scalar across all 400 h-slots of stack row 0.
#pragma unroll
    for (int m = 0; m < 16; ++m) {
        float sv = __shfl(sarr[m & 7], (m & 8) ? 16 : 0, 32);
        float* dst = out + (size_t)(t0 + m) * ROWF;   // row 0 of timestep t0+m
        for (int i = lane; i < HIDDEN; i += 32)
            __builtin_nontemporal_store(sv, dst + i);
    }
}

// ---------------------------------------------------------------------------
extern "C" void kernel_launch(void* const* d_in, const int* in_sizes, int n_in,
                              void* d_out, int out_size, void* d_ws, size_t ws_size,
                              hipStream_t stream) {
    const float* hs   = (const float*)d_in[0];  // (1, 8192, 400) f32
    const float* W    = (const float*)d_in[1];  // (3, 400) f32
    const float* bias = (const float*)d_in[2];  // (3,) f32
    const float* Dv   = (const float*)d_in[3];  // (1, 400) f32
    float* out = (float*)d_out;                 // (1, 8192, 32, 400) f32

    // 1) stream zeros into rows 1..31 (387.5 MB, the bandwidth-bound part)
    zero_fill_kernel<<<SEQ, 256, 0, stream>>>((v4f*)out);

    // 2) exact-f32 WMMA GEMV + activations + row-0 broadcast (512 waves)
    stack_wmma_kernel<<<128, 128, 0, stream>>>(hs, W, bias, Dv, out);
}